// GraphSAGE_Fraud_Detector_45432164057403
// MI455X (gfx1250) — compile-verified
//
#include <hip/hip_runtime.h>

// GraphSAGE 2-layer forward for MI455X (gfx1250, wave32).
// Bottleneck is the edge scatter (L2 fp32 atomics); GEMMs use exact fp32 WMMA
// (v_wmma_f32_16x16x4_f32) so no precision is lost vs the fp32 reference.

#define IN_C  128
#define HID_C 256
#define OUT_C 2

typedef __attribute__((ext_vector_type(2))) float v2f;
typedef __attribute__((ext_vector_type(8))) float v8f;

__device__ __forceinline__ void atomAddF32(float* p, float v) {
    (void)__hip_atomic_fetch_add(p, v, __ATOMIC_RELAXED, __HIP_MEMORY_SCOPE_AGENT);
}

__global__ void zero_f32(float* __restrict__ p, long n) {
    long i = (long)blockIdx.x * blockDim.x + threadIdx.x;
    long stride = (long)gridDim.x * blockDim.x;
    for (; i < n; i += stride) p[i] = 0.0f;
}

__global__ void degree_kernel(const int* __restrict__ dst, int E, float* __restrict__ cnt) {
    int e = blockIdx.x * blockDim.x + threadIdx.x;
    if (e < E) atomAddF32(cnt + dst[e], 1.0f);
}

// One thread per (edge, 4-channel chunk): float4 gather from feat[src], 4 f32
// atomics into agg[dst]. CHSHIFT = log2(C/4).
template <int C, int CHSHIFT>
__global__ void scatter_kernel(const float* __restrict__ feat,
                               const int* __restrict__ src,
                               const int* __restrict__ dst,
                               int E, float* __restrict__ agg) {
    long idx = (long)blockIdx.x * blockDim.x + threadIdx.x;
    const int CH = C / 4;                 // chunks per edge
    if (idx >= (long)E * CH) return;
    int e = (int)(idx >> CHSHIFT);
    int ch = (int)(idx & (CH - 1));
    int s = src[e], d = dst[e];
    const float4 vv = *(const float4*)(feat + (long)s * C + ch * 4);
    float* ap = agg + (long)d * C + ch * 4;
    atomAddF32(ap + 0, vv.x);
    atomAddF32(ap + 1, vv.y);
    atomAddF32(ap + 2, vv.z);
    atomAddF32(ap + 3, vv.w);
}

// Layer 1: h = relu((agg1/max(cnt,1)) @ W1l^T + b1 + x @ W1r^T)
// One wave per 16x16 output tile; fp32 WMMA, K = IN_C in steps of 4.
__global__ void __launch_bounds__(128)
gemm1_kernel(const float* __restrict__ agg1, const float* __restrict__ x,
             const float* __restrict__ cnt,
             const float* __restrict__ W1l, const float* __restrict__ b1,
             const float* __restrict__ W1r, float* __restrict__ h, int mTiles) {
    const int lane = threadIdx.x & 31;
    const int wave = threadIdx.x >> 5;
    const int nTiles = HID_C / 16;        // 16
    int tile = blockIdx.x * 4 + wave;
    if (tile >= mTiles * nTiles) return;  // wave-uniform exit: EXEC stays full
    const int mT = tile >> 4;
    const int nT = tile & 15;
    const int rc   = lane & 15;           // A-row within tile == B/D column
    const int half = lane >> 4;
    const int kkb  = half * 2;            // K sub-offset per ISA VGPR layout
    const int m0 = mT * 16, n0 = nT * 16;

    const float inv = 1.0f / fmaxf(cnt[m0 + rc], 1.0f);
    const float* Arow = agg1 + (long)(m0 + rc) * IN_C;
    const float* Xrow = x    + (long)(m0 + rc) * IN_C;
    const float* Bl   = W1l  + (long)(n0 + rc) * IN_C;  // B[k][n] = W[n][k]
    const float* Br   = W1r  + (long)(n0 + rc) * IN_C;

    v8f acc = {};
#pragma unroll 8
    for (int k0 = 0; k0 < IN_C; k0 += 4) {
        v2f a = *(const v2f*)(Arow + k0 + kkb);
        a *= inv;                                        // fold mean into A
        v2f b = *(const v2f*)(Bl + k0 + kkb);
        acc = __builtin_amdgcn_wmma_f32_16x16x4_f32(false, a, false, b,
                                                    (short)0, acc, false, false);
    }
#pragma unroll 8
    for (int k0 = 0; k0 < IN_C; k0 += 4) {
        v2f a = *(const v2f*)(Xrow + k0 + kkb);
        v2f b = *(const v2f*)(Br + k0 + kkb);
        acc = __builtin_amdgcn_wmma_f32_16x16x4_f32(false, a, false, b,
                                                    (short)0, acc, false, false);
    }
    const float bias = b1[n0 + rc];
#pragma unroll
    for (int v = 0; v < 8; ++v) {
        int m = m0 + v + 8 * half;                       // C/D VGPR layout
        h[(long)m * HID_C + n0 + rc] = fmaxf(acc[v] + bias, 0.0f);
    }
}

// Layer 2: out = (agg2/max(cnt,1)) @ W2l^T + b2 + h @ W2r^T   (only 2 cols live)
__global__ void __launch_bounds__(128)
gemm2_kernel(const float* __restrict__ agg2, const float* __restrict__ h,
             const float* __restrict__ cnt,
             const float* __restrict__ W2l, const float* __restrict__ b2,
             const float* __restrict__ W2r, float* __restrict__ out, int mTiles) {
    const int lane = threadIdx.x & 31;
    const int wave = threadIdx.x >> 5;
    int mT = blockIdx.x * 4 + wave;
    if (mT >= mTiles) return;
    const int rc   = lane & 15;
    const int half = lane >> 4;
    const int kkb  = half * 2;
    const int m0 = mT * 16;

    const float inv = 1.0f / fmaxf(cnt[m0 + rc], 1.0f);
    const float* Arow = agg2 + (long)(m0 + rc) * HID_C;
    const float* Hrow = h    + (long)(m0 + rc) * HID_C;
    const float* Bl = (rc < OUT_C) ? (W2l + (long)rc * HID_C) : nullptr;
    const float* Br = (rc < OUT_C) ? (W2r + (long)rc * HID_C) : nullptr;

    v8f acc = {};
#pragma unroll 8
    for (int k0 = 0; k0 < HID_C; k0 += 4) {
        v2f a = *(const v2f*)(Arow + k0 + kkb);
        a *= inv;
        v2f b = {0.0f, 0.0f};
        if (rc < OUT_C) b = *(const v2f*)(Bl + k0 + kkb);  // divergent load only
        acc = __builtin_amdgcn_wmma_f32_16x16x4_f32(false, a, false, b,
                                                    (short)0, acc, false, false);
    }
#pragma unroll 8
    for (int k0 = 0; k0 < HID_C; k0 += 4) {
        v2f a = *(const v2f*)(Hrow + k0 + kkb);
        v2f b = {0.0f, 0.0f};
        if (rc < OUT_C) b = *(const v2f*)(Br + k0 + kkb);
        acc = __builtin_amdgcn_wmma_f32_16x16x4_f32(false, a, false, b,
                                                    (short)0, acc, false, false);
    }
    if (rc < OUT_C) {
        const float bias = b2[rc];
#pragma unroll
        for (int v = 0; v < 8; ++v) {
            int m = m0 + v + 8 * half;
            out[(long)m * OUT_C + rc] = acc[v] + bias;
        }
    }
}

extern "C" void kernel_launch(void* const* d_in, const int* in_sizes, int n_in,
                              void* d_out, int out_size, void* d_ws, size_t ws_size,
                              hipStream_t stream) {
    const float* x   = (const float*)d_in[0];
    const int*   ei  = (const int*)  d_in[1];
    const float* W1l = (const float*)d_in[2];
    const float* b1  = (const float*)d_in[3];
    const float* W1r = (const float*)d_in[4];
    const float* W2l = (const float*)d_in[5];
    const float* b2  = (const float*)d_in[6];
    const float* W2r = (const float*)d_in[7];
    float* out = (float*)d_out;

    const int N = in_sizes[0] / IN_C;     // 50000 (multiple of 16)
    const int E = in_sizes[1] / 2;        // 800000
    const int* src = ei;
    const int* dst = ei + E;

    // Workspace layout: cnt | agg1 | h | agg2  (~128 MB)
    const long padN = ((long)N + 63) & ~63L;
    float* cnt  = (float*)d_ws;
    float* agg1 = cnt  + padN;
    float* h    = agg1 + (long)N * IN_C;
    float* agg2 = h    + (long)N * HID_C;

    // Re-zero accumulators every call (graph replays must be deterministic).
    zero_f32<<<4096, 256, 0, stream>>>(cnt,  padN + (long)N * IN_C);
    zero_f32<<<4096, 256, 0, stream>>>(agg2, (long)N * HID_C);

    degree_kernel<<<(E + 255) / 256, 256, 0, stream>>>(dst, E, cnt);

    {   // x[src] -> agg1[dst], 128 channels, 32 chunks/edge
        long total = (long)E * (IN_C / 4);
        scatter_kernel<IN_C, 5><<<(int)((total + 255) / 256), 256, 0, stream>>>(
            x, src, dst, E, agg1);
    }

    const int mTiles = N / 16;            // 3125
    gemm1_kernel<<<(mTiles * (HID_C / 16) + 3) / 4, 128, 0, stream>>>(
        agg1, x, cnt, W1l, b1, W1r, h, mTiles);

    {   // h[src] -> agg2[dst], 256 channels, 64 chunks/edge
        long total = (long)E * (HID_C / 4);
        scatter_kernel<HID_C, 6><<<(int)((total + 255) / 256), 256, 0, stream>>>(
            h, src, dst, E, agg2);
    }

    gemm2_kernel<<<(mTiles + 3) / 4, 128, 0, stream>>>(
        agg2, h, cnt, W2l, b2, W2r, out, mTiles);
}